// ChamferLoss_45157286150461
// MI455X (gfx1250) — compile-verified
//
#include <hip/hip_runtime.h>
#include <hip/hip_bf16.h>

typedef __attribute__((ext_vector_type(2))) float v2f;
typedef __attribute__((ext_vector_type(8))) float v8f;

// Zero the single-float accumulator (harness poisons d_out with 0xAA).
__global__ void chamfer_zero_kernel(float* out) { out[0] = 0.0f; }

// Both directed Chamfer passes in one grid (blockIdx.y selects direction).
// For each column point c, compute min over all row points r of
//   ||r||^2 + ||c||^2 - 2 r.c
// and atomically accumulate scale * sum(min) into out[0].
//
// V_WMMA_F32_16X16X4_F32 with the K=4 slot carrying the row norm:
//   A[m][0..2] = row coords,      A[m][3] = ||row_m||^2
//   B[0..2][n] = -2 * col coords, B[3][n] = 1.0
// => (A x B)[m][n] = ||row_m||^2 - 2 row_m . col_n   (add ||col_n||^2 post-min)
//
// Wave layout (wave32):
//   A 16x4 f32: lanes 0-15 hold M=lane {K0,K1}; lanes 16-31 hold M=lane-16 {K2,K3}
//   B 4x16 f32: lanes 0-15 hold N=lane {K0,K1}; lanes 16-31 hold N=lane-16 {K2,K3}
//   C/D 16x16 : VGPR v -> M=v (lanes 0-15) / M=v+8 (lanes 16-31), N=lane&15
//
// Column blocking: each wave owns CT=4 column tiles (64 columns) and reuses one
// A fragment across 4 WMMAs per row-tile iteration.
#define CT 4

__global__ __launch_bounds__(256) void chamfer_pass_kernel(
    const float* __restrict__ preds,   // [B, N, 3]
    const float* __restrict__ gts,     // [B, N, 3]
    float* __restrict__ out,
    int nPts, float scale)
{
    // Direction: y==0 -> dl (cols = preds, min over gts rows)
    //            y==1 -> dr (cols = gts,   min over preds rows)
    const float* __restrict__ rowPts = (blockIdx.y == 0) ? gts   : preds;
    const float* __restrict__ colPts = (blockIdx.y == 0) ? preds : gts;

    const int lane    = threadIdx.x & 31;
    const int wave    = threadIdx.x >> 5;
    const int l15     = lane & 15;
    const bool hiHalf = (lane >= 16);

    // 8 waves/block, 64 columns/wave -> 512 columns/block.
    const int blocksPerBatch = nPts / (8 * 16 * CT);       // 8 for N=4096
    const int b   = blockIdx.x / blocksPerBatch;
    const int grp = blockIdx.x % blocksPerBatch;
    const int colBase = grp * (8 * 16 * CT) + wave * (16 * CT) + l15;

    const size_t batchBase = (size_t)b * (size_t)nPts * 3u;

    // ---- Build CT B fragments (columns) once, plus per-lane column norms. ----
    v2f   bfrag[CT];
    float yy[CT];
    float colmin[CT];
#pragma unroll
    for (int t = 0; t < CT; ++t) {
        const int col = colBase + t * 16;
        const float* cp = colPts + batchBase + (size_t)col * 3u;
        const float c0 = cp[0], c1 = cp[1], c2 = cp[2];
        yy[t] = c0 * c0 + c1 * c1 + c2 * c2;
        bfrag[t].x = hiHalf ? (-2.0f * c2) : (-2.0f * c0); // K2 : K0
        bfrag[t].y = hiHalf ? 1.0f         : (-2.0f * c1); // K3 : K1
        colmin[t]  = 3.4028235e38f;
    }

    for (int rt = 0; rt < nPts; rt += 16) {
        // ---- A fragment: 16 row points; lane loads point rt + (lane&15). ----
        const float* rp = rowPts + batchBase + (size_t)(rt + l15) * 3u;
        const float a0 = rp[0], a1 = rp[1], a2 = rp[2];
        const float xx = a0 * a0 + a1 * a1 + a2 * a2;
        v2f afrag;
        afrag.x = hiHalf ? a2 : a0;                        // K2 : K0
        afrag.y = hiHalf ? xx : a1;                        // K3 : K1

#pragma unroll
        for (int t = 0; t < CT; ++t) {
            v8f c = {};
            v8f d = __builtin_amdgcn_wmma_f32_16x16x4_f32(
                /*neg_a=*/false, afrag, /*neg_b=*/false, bfrag[t],
                /*c_mod=*/(short)0, c, /*reuse_a=*/false, /*reuse_b=*/false);

            // Column-min epilogue: all 8 D regs belong to this lane's column.
            float m = colmin[t];
            m = fminf(m, d[0]); m = fminf(m, d[1]);
            m = fminf(m, d[2]); m = fminf(m, d[3]);
            m = fminf(m, d[4]); m = fminf(m, d[5]);
            m = fminf(m, d[6]); m = fminf(m, d[7]);
            colmin[t] = m;
        }
    }

    // Lane L holds min over tile-rows 0..7, lane L+16 over rows 8..15: combine,
    // add column norm, and sum this lane's CT columns.
    float acc = 0.0f;
#pragma unroll
    for (int t = 0; t < CT; ++t) {
        float m = fminf(colmin[t], __shfl_xor(colmin[t], 16, 32));
        acc += m + yy[t];
    }

    // Sum the 16 distinct low-lane contributions (hi lanes are duplicates).
    float v = hiHalf ? 0.0f : acc;
    v += __shfl_xor(v, 1, 32);
    v += __shfl_xor(v, 2, 32);
    v += __shfl_xor(v, 4, 32);
    v += __shfl_xor(v, 8, 32);
    v += __shfl_xor(v, 16, 32);

    if (lane == 0) {
        atomicAdd(out, v * scale);
    }
}

extern "C" void kernel_launch(void* const* d_in, const int* in_sizes, int n_in,
                              void* d_out, int out_size, void* d_ws, size_t ws_size,
                              hipStream_t stream) {
    const float* preds = (const float*)d_in[0];   // [B, N, 3]
    const float* gts   = (const float*)d_in[1];   // [B, N, 3]
    float* out = (float*)d_out;

    const int N = 4096;
    const int B = in_sizes[0] / (N * 3);
    const float scale = 1.0f / ((float)B * (float)N);

    chamfer_zero_kernel<<<1, 1, 0, stream>>>(out);

    // x: (batch, column-group); y: direction (dl, dr)
    dim3 grid(B * (N / (8 * 16 * CT)), 2);
    chamfer_pass_kernel<<<grid, 256, 0, stream>>>(preds, gts, out, N, scale);
}